// NPStatePrior_39075612459507
// MI455X (gfx1250) — compile-verified
//
#include <hip/hip_runtime.h>
#include <cmath>

typedef float v2f __attribute__((ext_vector_type(2)));
typedef float v8f __attribute__((ext_vector_type(8)));

#define SLOPE 0.2f
#define NSAMP 65536
#define D_DIM 8
#define T_LEN 1024
#define WAVES 4
#define ROWS_PER_BLOCK (WAVES * 16)
#define WSTRIDE 68            // 64 + 4 pad: conflict-free b64 LDS loads for WMMA operands
#define RES_ELEMS (NSAMP * D_DIM)

__global__ void zero_logdet_kernel(float* __restrict__ out) {
    out[RES_ELEMS + threadIdx.x] = 0.0f;   // 64 per-batch logdet accumulators
}

// One fused pass: forward MLP + analytic tangent, fp32 WMMA 16x16x4.
__device__ __forceinline__ void gemm_pair_64x64(
    const float* __restrict__ aLds, const float* __restrict__ tLds,
    const float* __restrict__ wLds, int m, int khalf,
    v8f accF[4], v8f accT[4])
{
    const v8f vz = {};
#pragma unroll
    for (int jn = 0; jn < 4; ++jn) { accF[jn] = vz; accT[jn] = vz; }
#pragma unroll
    for (int kk = 0; kk < 16; ++kk) {
        const int kb = kk * 4 + khalf;
        v2f af; af.x = aLds[m * WSTRIDE + kb]; af.y = aLds[m * WSTRIDE + kb + 1];
        v2f at; at.x = tLds[m * WSTRIDE + kb]; at.y = tLds[m * WSTRIDE + kb + 1];
#pragma unroll
        for (int jn = 0; jn < 4; ++jn) {
            v2f bb;
            bb.x = wLds[(jn * 16 + m) * WSTRIDE + kb];
            bb.y = wLds[(jn * 16 + m) * WSTRIDE + kb + 1];
            accF[jn] = __builtin_amdgcn_wmma_f32_16x16x4_f32(
                false, af, false, bb, (short)0, accF[jn], false, false);
            accT[jn] = __builtin_amdgcn_wmma_f32_16x16x4_f32(
                false, at, false, bb, (short)0, accT[jn], false, false);
        }
    }
}

__global__ __launch_bounds__(WAVES * 32) void flow_mlp_kernel(
    const float* __restrict__ z,  const float* __restrict__ xs,
    const float* __restrict__ W1, const float* __restrict__ b1,
    const float* __restrict__ W2, const float* __restrict__ b2,
    const float* __restrict__ W3, const float* __restrict__ b3,
    const float* __restrict__ Wout, const float* __restrict__ bout,
    float* __restrict__ out)
{
    __shared__ float W1lds[64 * 12];          // K padded 9 -> 12
    __shared__ float wbuf[64 * WSTRIDE];      // holds W2, then W3
    __shared__ float b1l[64], b2l[64], b3l[64], woutl[64];
    __shared__ float inpT[WAVES][16 * 12];
    __shared__ float aT[WAVES][16 * WSTRIDE];
    __shared__ float tT[WAVES][16 * WSTRIDE];

    const int tid   = threadIdx.x;
    const int lane  = tid & 31;
    const int wave  = tid >> 5;
    const int m     = lane & 15;     // A-row / B-col within 16-tile
    const int hi    = lane >> 4;     // wave half (C/D M-split)
    const int khalf = hi << 1;       // k offset within 4-wide chunk: 0 or 2
    const int d     = blockIdx.y;
    const int n0w   = blockIdx.x * ROWS_PER_BLOCK + wave * 16;

    // ---- cooperative staging: W1 (zero-padded), biases, Wout, W2 ----
    for (int idx = tid; idx < 64 * 12; idx += WAVES * 32) {
        const int h = idx / 12, c = idx % 12;
        W1lds[idx] = (c < 9) ? W1[(d * 64 + h) * 9 + c] : 0.0f;
    }
    if (tid < 64) {
        b1l[tid]   = b1[d * 64 + tid];
        b2l[tid]   = b2[d * 64 + tid];
        b3l[tid]   = b3[d * 64 + tid];
        woutl[tid] = Wout[d * 64 + tid];
    }
    for (int idx = tid; idx < 64 * 64; idx += WAVES * 32) {
        const int g = idx >> 6, h = idx & 63;
        wbuf[g * WSTRIDE + h] = W2[(d * 64 + g) * 64 + h];
    }
    // per-wave input tile: cols 0..7 = z row, col 8 = xs[:,d], 9..11 = 0
    float* ip = inpT[wave];
    for (int idx = lane; idx < 16 * 12; idx += 32) {
        const int row = idx / 12, c = idx % 12;
        const int n = n0w + row;
        float v = 0.0f;
        if (c < 8)       v = z[n * 8 + c];
        else if (c == 8) v = xs[n * 8 + d];
        ip[idx] = v;
    }
    __syncthreads();

    // ---- layer 1: pre1 = inp(16x12) * W1^T  (WMMA, K padded) ----
    v8f accF[4], accT[4];
    {
        const v8f vz = {};
#pragma unroll
        for (int jn = 0; jn < 4; ++jn) accF[jn] = vz;
#pragma unroll
        for (int kk = 0; kk < 3; ++kk) {
            const int kb = kk * 4 + khalf;
            v2f a; a.x = ip[m * 12 + kb]; a.y = ip[m * 12 + kb + 1];
#pragma unroll
            for (int jn = 0; jn < 4; ++jn) {
                v2f bb;
                bb.x = W1lds[(jn * 16 + m) * 12 + kb];
                bb.y = W1lds[(jn * 16 + m) * 12 + kb + 1];
                accF[jn] = __builtin_amdgcn_wmma_f32_16x16x4_f32(
                    false, a, false, bb, (short)0, accF[jn], false, false);
            }
        }
    }
    float* aw = aT[wave];
    float* tw = tT[wave];
#pragma unroll
    for (int jn = 0; jn < 4; ++jn) {
        const int h = jn * 16 + m;
        const float bh  = b1l[h];
        const float w1l = W1lds[h * 12 + 8];      // W1[:, -1]
#pragma unroll
        for (int r = 0; r < 8; ++r) {
            const float p  = accF[jn][r] + bh;
            const float dl = (p >= 0.0f) ? 1.0f : SLOPE;
            const int mm = r + (hi << 3);
            aw[mm * WSTRIDE + h] = p * dl;        // leaky(pre1)
            tw[mm * WSTRIDE + h] = w1l * dl;      // t1
        }
    }
    __syncthreads();

    // ---- layer 2: forward + tangent share W2 B-operands ----
    gemm_pair_64x64(aw, tw, wbuf, m, khalf, accF, accT);
#pragma unroll
    for (int jn = 0; jn < 4; ++jn) {
        const int h = jn * 16 + m;
        const float bh = b2l[h];
#pragma unroll
        for (int r = 0; r < 8; ++r) {
            const float p  = accF[jn][r] + bh;
            const float dl = (p >= 0.0f) ? 1.0f : SLOPE;
            const int mm = r + (hi << 3);
            aw[mm * WSTRIDE + h] = p * dl;
            tw[mm * WSTRIDE + h] = accT[jn][r] * dl;
        }
    }
    __syncthreads();                 // everyone done with W2 + tiles written
    for (int idx = tid; idx < 64 * 64; idx += WAVES * 32) {
        const int g = idx >> 6, h = idx & 63;
        wbuf[g * WSTRIDE + h] = W3[(d * 64 + g) * 64 + h];
    }
    __syncthreads();

    // ---- layer 3 ----
    gemm_pair_64x64(aw, tw, wbuf, m, khalf, accF, accT);
#pragma unroll
    for (int jn = 0; jn < 4; ++jn) {
        const int h = jn * 16 + m;
        const float bh = b3l[h];
#pragma unroll
        for (int r = 0; r < 8; ++r) {
            const float p  = accF[jn][r] + bh;
            const float dl = (p >= 0.0f) ? 1.0f : SLOPE;
            accF[jn][r] = p * dl;                 // a3
            accT[jn][r] = accT[jn][r] * dl;       // t3
        }
    }

    // ---- output contraction with Wout + cross-lane row reduction ----
    float wo[4];
#pragma unroll
    for (int jn = 0; jn < 4; ++jn) wo[jn] = woutl[jn * 16 + m];
    const float bo = bout[d];

    float so[8], st[8];
#pragma unroll
    for (int r = 0; r < 8; ++r) {
        float s0 = 0.0f, s1 = 0.0f;
#pragma unroll
        for (int jn = 0; jn < 4; ++jn) {
            s0 += accF[jn][r] * wo[jn];
            s1 += accT[jn][r] * wo[jn];
        }
#pragma unroll
        for (int off = 1; off < 16; off <<= 1) {  // reduce 16 lanes of each half
            s0 += __shfl_xor(s0, off, 32);
            s1 += __shfl_xor(s1, off, 32);
        }
        so[r] = s0; st[r] = s1;
    }

    if (m == 0) {                                 // lanes 0 and 16
        const int mbase = hi << 3;
        float lacc = 0.0f;
#pragma unroll
        for (int r = 0; r < 8; ++r) {
            const int n = n0w + mbase + r;
            out[n * D_DIM + d] = so[r] + bo;      // residuals (B,T,D)
            lacc += logf(fabsf(st[r]));
        }
        atomicAdd(&out[RES_ELEMS + (n0w >> 10)], lacc);  // per-b logdet
    }
}

extern "C" void kernel_launch(void* const* d_in, const int* in_sizes, int n_in,
                              void* d_out, int out_size, void* d_ws, size_t ws_size,
                              hipStream_t stream) {
    (void)in_sizes; (void)n_in; (void)out_size; (void)d_ws; (void)ws_size;
    const float* z    = (const float*)d_in[0];
    const float* xs   = (const float*)d_in[1];
    const float* W1   = (const float*)d_in[2];
    const float* b1   = (const float*)d_in[3];
    const float* W2   = (const float*)d_in[4];
    const float* b2   = (const float*)d_in[5];
    const float* W3   = (const float*)d_in[6];
    const float* b3   = (const float*)d_in[7];
    const float* Wout = (const float*)d_in[8];
    const float* bout = (const float*)d_in[9];
    float* out = (float*)d_out;

    zero_logdet_kernel<<<1, 64, 0, stream>>>(out);
    dim3 grid(NSAMP / ROWS_PER_BLOCK, D_DIM);
    flow_mlp_kernel<<<grid, WAVES * 32, 0, stream>>>(
        z, xs, W1, b1, W2, b2, W3, b3, Wout, bout, out);
}